// NCykParser_9405978378690
// MI455X (gfx1250) — compile-verified
//
#include <hip/hip_runtime.h>
#include <float.h>

// CYK max-product parser for MI455X (gfx1250, wave32).
// (max,x)-semiring: WMMA is inapplicable (it computes sum-of-products only).
// CDNA5 paths used: global_load_async_to_lds_b128 (ASYNCcnt) with double
// buffering, s_wait_asynccnt, global_prefetch_b8, wave32 shfl_xor reductions.
// chart stored triangularly in d_ws: cell (L,i) at ((194-L)(L-1)/2 + i)*64 floats.

#define N_SEQ 96
#define R 64
#define KC 8  // splits staged per LDS chunk

#if defined(__has_builtin)
#if __has_builtin(__builtin_amdgcn_global_load_async_to_lds_b128) && \
    __has_builtin(__builtin_amdgcn_s_wait_asynccnt)
#define USE_ASYNC_LDS 1
#endif
#endif
#ifndef USE_ASYNC_LDS
#define USE_ASYNC_LDS 0
#endif

#if USE_ASYNC_LDS
// Exact param type from clang diagnostic: 'int __vector(4) __device__ *'
typedef int v4i_t __attribute__((vector_size(16)));
typedef v4i_t __attribute__((address_space(1))) * g_b128_t;  // global (AS1)
typedef v4i_t __attribute__((address_space(3))) * l_b128_t;  // LDS (AS3)
#endif

__device__ __forceinline__ int cell_base(int L, int i) {
  // triangular layout: sum of span counts for lengths < L, plus i; in floats
  return ((194 - L) * (L - 1) / 2 + i) * R;
}

// Stage U=chart(i,i+k), V=chart(i+k,i+L) for k=kb..kb+kcnt-1 into one UV buffer.
// Wave w handles split kb+w: lanes 0-15 copy U (16B/lane), 16-31 copy V.
__device__ __forceinline__ void stage_chunk(const float* __restrict__ chart,
                                            float (*UVb)[2][R], int i, int L,
                                            int kb, int wave, int lane) {
  const int kcnt = min(KC, L - kb);
  if (wave < kcnt) {
    const int k = kb + wave;
    const float* src;
    float* dst;
    if (lane < 16) {
      src = chart + cell_base(k, i) + lane * 4;  // left child, length k
      dst = &UVb[wave][0][lane * 4];
    } else {
      src = chart + cell_base(L - k, i + k) + (lane - 16) * 4;  // right child
      dst = &UVb[wave][1][(lane - 16) * 4];
    }
#if USE_ASYNC_LDS
    __builtin_amdgcn_global_load_async_to_lds_b128((g_b128_t)src, (l_b128_t)dst,
                                                   0, 0);
#else
    *(float4*)dst = *(const float4*)src;
#endif
  }
}

__global__ __launch_bounds__(256) void cyk_step(const float* __restrict__ W,
                                                float* __restrict__ chart,
                                                float* __restrict__ out,
                                                int L) {
  __shared__ __align__(16) float UV[2][KC][2][R];  // double-buffered staging (8 KB)
  __shared__ __align__(16) float PM[2][R * R];     // Pmax/Pmin (32 KB)

  const int i = blockIdx.x;  // span start; span covers s[i : i+L]
  const int tid = (int)threadIdx.x;
  const int wave = tid >> 5;
  const int lane = tid & 31;

  // ---- Phase 1: Pmax/Pmin over splits k. Thread owns 4x4 (a,b) tile. ----
  const int a0 = (tid >> 4) << 2;  // 16 a-groups of 4
  const int b0 = (tid & 15) << 2;  // 16 b-groups of 4
  float pmx[16], pmn[16];
#pragma unroll
  for (int t = 0; t < 16; ++t) {
    pmx[t] = -FLT_MAX;
    pmn[t] = FLT_MAX;
  }

  int p = 0;
  stage_chunk(chart, UV[0], i, L, 1, wave, lane);  // prime buffer 0
  for (int kb = 1; kb < L; kb += KC) {
    const int kcnt = min(KC, L - kb);
#if USE_ASYNC_LDS
    __builtin_amdgcn_s_wait_asynccnt(0);  // this wave's staged copies landed
#endif
    __syncthreads();  // all waves' copies for UV[p] visible; UV[p^1] free

    if (kb + KC < L)  // prefetch next chunk while computing this one
      stage_chunk(chart, UV[p ^ 1], i, L, kb + KC, wave, lane);

    for (int kk = 0; kk < kcnt; ++kk) {
      const float4 u4 = *(const float4*)&UV[p][kk][0][a0];  // ds_load_b128
      const float4 v4 = *(const float4*)&UV[p][kk][1][b0];  // ds_load_b128
      const float u[4] = {u4.x, u4.y, u4.z, u4.w};
      const float v[4] = {v4.x, v4.y, v4.z, v4.w};
#pragma unroll
      for (int x = 0; x < 4; ++x)
#pragma unroll
        for (int y = 0; y < 4; ++y) {
          const float o = u[x] * v[y];
          const int t = x * 4 + y;
          pmx[t] = fmaxf(pmx[t], o);
          pmn[t] = fminf(pmn[t], o);
        }
    }
    __syncthreads();  // everyone done reading UV[p] before it is restaged
    p ^= 1;
  }

  // Publish tile to LDS (b128 stores: rows of 4 consecutive b).
#pragma unroll
  for (int x = 0; x < 4; ++x) {
    const int e = (a0 + x) * R + b0;
    *(float4*)&PM[0][e] = make_float4(pmx[x * 4 + 0], pmx[x * 4 + 1],
                                      pmx[x * 4 + 2], pmx[x * 4 + 3]);
    *(float4*)&PM[1][e] = make_float4(pmn[x * 4 + 0], pmn[x * 4 + 1],
                                      pmn[x * 4 + 2], pmn[x * 4 + 3]);
  }
  __syncthreads();

  // ---- Phase 2: result[j] = max(0, max_{a,b} max(W*Pmax, W*Pmin)). ----
  // Wave w owns rules j = 8w..8w+7. Lanes sweep the 4096 (a,b) cells in
  // float4 strides (coalesced global_load_b128 on W, conflict-free
  // ds_load_b128 on PM, PM reads hoisted across the 8 rules), then a
  // wave32 shfl_xor max-reduction per rule.
  const float4* __restrict__ PM0 = (const float4*)&PM[0][0];
  const float4* __restrict__ PM1 = (const float4*)&PM[1][0];
  const float* Wbase = W + (wave * 8) * (R * R);
  __builtin_prefetch(Wbase, 0, 1);  // global_prefetch_b8 (W is L2-resident)

  float4 acc[8];
#pragma unroll
  for (int jj = 0; jj < 8; ++jj)
    acc[jj] = make_float4(-FLT_MAX, -FLT_MAX, -FLT_MAX, -FLT_MAX);

  for (int it = 0; it < (R * R) / (32 * 4); ++it) {  // 32 iterations
    const int e4 = it * 32 + lane;                   // float4 index in (a,b) plane
    const float4 px = PM0[e4];
    const float4 pn = PM1[e4];
#pragma unroll
    for (int jj = 0; jj < 8; ++jj) {
      const float4 w4 = ((const float4*)(Wbase + jj * (R * R)))[e4];
      acc[jj].x = fmaxf(acc[jj].x, fmaxf(w4.x * px.x, w4.x * pn.x));
      acc[jj].y = fmaxf(acc[jj].y, fmaxf(w4.y * px.y, w4.y * pn.y));
      acc[jj].z = fmaxf(acc[jj].z, fmaxf(w4.z * px.z, w4.z * pn.z));
      acc[jj].w = fmaxf(acc[jj].w, fmaxf(w4.w * px.w, w4.w * pn.w));
    }
  }

  float* dstcell = chart + cell_base(L, i);
#pragma unroll
  for (int jj = 0; jj < 8; ++jj) {
    float m = fmaxf(fmaxf(acc[jj].x, acc[jj].y), fmaxf(acc[jj].z, acc[jj].w));
#pragma unroll
    for (int off = 16; off > 0; off >>= 1) m = fmaxf(m, __shfl_xor(m, off, 32));
    if (lane == 0) {
      const float rj = fmaxf(m, 0.0f);
      dstcell[wave * 8 + jj] = rj;
      if (L == N_SEQ && i == 0 && wave == 0 && jj == 0) out[0] = rj;  // chart[0,n,0]
    }
  }
}

__global__ void cyk_init(const int* __restrict__ tokens,
                         const float* __restrict__ E,
                         float* __restrict__ chart) {
  int idx = blockIdx.x * blockDim.x + threadIdx.x;
  if (idx >= N_SEQ * R) return;
  int i = idx >> 6;
  int r = idx & (R - 1);
  chart[cell_base(1, i) + r] = E[tokens[i] * R + r];
}

extern "C" void kernel_launch(void* const* d_in, const int* in_sizes, int n_in,
                              void* d_out, int out_size, void* d_ws, size_t ws_size,
                              hipStream_t stream) {
  (void)in_sizes; (void)n_in; (void)out_size; (void)ws_size;
  const int* tokens = (const int*)d_in[0];  // [96] int32
  const float* W = (const float*)d_in[1];   // [64,64,64] f32
  const float* E = (const float*)d_in[2];   // [32,64] f32
  float* out = (float*)d_out;               // [1] f32
  float* chart = (float*)d_ws;              // 4656 cells * 64 f32 = ~1.2 MB

  cyk_init<<<(N_SEQ * R + 255) / 256, 256, 0, stream>>>(tokens, E, chart);
  for (int L = 2; L <= N_SEQ; ++L) {
    const int spans = N_SEQ - L + 1;
    cyk_step<<<spans, 256, 0, stream>>>(W, chart, out, L);
  }
}